// SpERTRelationClassifier_20134806684054
// MI455X (gfx1250) — compile-verified
//
#include <hip/hip_runtime.h>
#include <hip/hip_bf16.h>

// ---------------------------------------------------------------------------
// SpERT relation classifier head for MI455X (gfx1250, wave32, WMMA).
//
// Shapes: B=256, S=512, H=768, WDIM=25, NLAB=42, CIN=2354 (padded to 2368).
// Pipeline:
//   1) build_combined : gather e1/e2 reps, width embeddings, masked-mean ctx
//                       -> combined [256, 2368] (zero padded K tail)   (HBM bound)
//   2) gemm1_relu     : h = relu(combined @ W1 + b1) via V_WMMA_F32_16X16X4_F32
//   3) gemm2          : logits = h @ W2 + b2 via V_WMMA_F32_16X16X4_F32
// ---------------------------------------------------------------------------

typedef __attribute__((ext_vector_type(2))) float v2f;
typedef __attribute__((ext_vector_type(8))) float v8f;

#define B_SZ   256
#define S_SZ   512
#define H_SZ   768
#define WDIM   25
#define NLAB   42
#define CIN    2354
#define CINP   2368          // CIN padded to multiple of 8 (zero filled)
#define KMAIN  2352          // largest multiple of 8 <= CIN
#define MAXW   10

// ---------------------------------------------------------------------------
// Kernel 1: build combined feature matrix [B, CINP] (row-major, zero padded).
// Layout per row: [0,768) e1_repr | [768,1536) e2_repr | [1536,1561) e1_w |
//                 [1561,1586) e2_w | [1586,2354) ctx | [2354,2368) zeros
// One block (256 threads) per batch element; 3 H-elements per thread.
// ---------------------------------------------------------------------------
__global__ __launch_bounds__(256)
void build_combined(const float* __restrict__ seq,
                    const int* __restrict__ e1p, const int* __restrict__ e2p,
                    const int* __restrict__ l1,  const int* __restrict__ l2,
                    const float* __restrict__ wemb,
                    float* __restrict__ comb)
{
    const int b = blockIdx.x;
    const int t = threadIdx.x;

    const int e1 = e1p[b];
    const int e2 = e2p[b];

    const float* base = seq + (size_t)b * S_SZ * H_SZ;
    float* out = comb + (size_t)b * CINP;

    // masked mean over tokens strictly between e1 and e2
    float a0 = 0.f, a1 = 0.f, a2 = 0.f;
    const int lo = e1 + 1, hi = e2;
    for (int s = lo; s < hi; ++s) {
        const float* r = base + (size_t)s * H_SZ;
        a0 += r[t];
        a1 += r[t + 256];
        a2 += r[t + 512];
    }
    const int cnt = hi - lo;
    const float inv = (cnt > 0) ? (1.0f / (float)cnt) : 0.0f;

    // entity token representations
    const float* r1 = base + (size_t)e1 * H_SZ;
    const float* r2 = base + (size_t)e2 * H_SZ;
    out[t]        = r1[t];
    out[t + 256]  = r1[t + 256];
    out[t + 512]  = r1[t + 512];
    out[768 + t]        = r2[t];
    out[768 + t + 256]  = r2[t + 256];
    out[768 + t + 512]  = r2[t + 512];

    // context (masked mean, zero when empty span)
    out[1586 + t]        = a0 * inv;
    out[1586 + t + 256]  = a1 * inv;
    out[1586 + t + 512]  = a2 * inv;

    // width embeddings (clamped to MAX_SPAN_WIDTH)
    if (t < WDIM) {
        int i1 = l1[b]; i1 = i1 < 0 ? 0 : (i1 > MAXW ? MAXW : i1);
        int i2 = l2[b]; i2 = i2 < 0 ? 0 : (i2 > MAXW ? MAXW : i2);
        out[1536 + t] = wemb[i1 * WDIM + t];
        out[1561 + t] = wemb[i2 * WDIM + t];
    }
    // zero pad K tail so WMMA K-tiles past CIN contribute nothing
    if (t < CINP - CIN) out[CIN + t] = 0.0f;
}

// ---------------------------------------------------------------------------
// Kernel 2: h = relu(combined @ W1 + b1), WMMA f32 16x16x4.
// One wave32 per 16x16 output tile; 16 row tiles x 48 col tiles = 768 waves.
// f32 WMMA operand layout:
//   A (16x4): lane M=lane&15 holds {K=kOff, K=kOff+1}, kOff = 2*(lane>>4)
//   B (4x16): lane N=lane&15 holds {K=kOff, K=kOff+1}
//   C/D     : vgpr r -> (M = 8*(lane>>4) + r, N = lane&15)
// Two accumulators on interleaved K-steps hide the WMMA dependency chain.
// ---------------------------------------------------------------------------
__global__ __launch_bounds__(256)
void gemm1_relu(const float* __restrict__ A, const float* __restrict__ W1,
                const float* __restrict__ bias1, float* __restrict__ Hout)
{
    const int gid  = blockIdx.x * blockDim.x + threadIdx.x;
    const int wave = gid >> 5;
    const int lane = threadIdx.x & 31;
    const int tm = wave / (H_SZ / 16);       // 0..15
    const int tn = wave % (H_SZ / 16);       // 0..47
    const int m0 = tm * 16, n0 = tn * 16;
    const int half = lane >> 4;              // 0 or 1
    const int l16  = lane & 15;
    const int kOff = half * 2;               // 0 or 2

    const float* Ap = A  + (size_t)(m0 + l16) * CINP + kOff;   // per-lane A row
    const float* Bp = W1 + (size_t)kOff * H_SZ + (n0 + l16);   // per-lane B col

    v8f c0 = {}; v8f c1 = {};
    #pragma unroll 2
    for (int k = 0; k < KMAIN; k += 8) {
        v2f a0 = *(const v2f*)(Ap + k);
        v2f b0; b0.x = Bp[k * H_SZ];       b0.y = Bp[(k + 1) * H_SZ];
        c0 = __builtin_amdgcn_wmma_f32_16x16x4_f32(false, a0, false, b0,
                                                   (short)0, c0, false, false);
        v2f a1 = *(const v2f*)(Ap + k + 4);
        v2f b1; b1.x = Bp[(k + 4) * H_SZ]; b1.y = Bp[(k + 5) * H_SZ];
        c1 = __builtin_amdgcn_wmma_f32_16x16x4_f32(false, a1, false, b1,
                                                   (short)0, c1, false, false);
    }
    // peeled tail: K = 2352..2355. A cols >= 2354 are zero padded; mask W1 rows.
    {
        v2f a0 = *(const v2f*)(Ap + KMAIN);
        v2f b0;
        b0.x = (KMAIN + kOff     < CIN) ? Bp[KMAIN * H_SZ]       : 0.0f;
        b0.y = (KMAIN + kOff + 1 < CIN) ? Bp[(KMAIN + 1) * H_SZ] : 0.0f;
        c0 = __builtin_amdgcn_wmma_f32_16x16x4_f32(false, a0, false, b0,
                                                   (short)0, c0, false, false);
    }

    const int   ncol  = n0 + l16;
    const float bv    = bias1[ncol];
    const int   rbase = m0 + half * 8;
    #pragma unroll
    for (int r = 0; r < 8; ++r) {
        float v = c0[r] + c1[r] + bv;
        Hout[(size_t)(rbase + r) * H_SZ + ncol] = v > 0.0f ? v : 0.0f;
    }
}

// ---------------------------------------------------------------------------
// Kernel 3: logits = h @ W2 + b2, WMMA f32 16x16x4.
// N=42 padded to 48 -> 16 row tiles x 3 col tiles = 48 waves. K=768 (no tail).
// ---------------------------------------------------------------------------
__global__ __launch_bounds__(128)
void gemm2(const float* __restrict__ Hin, const float* __restrict__ W2,
           const float* __restrict__ bias2, float* __restrict__ Out)
{
    const int gid  = blockIdx.x * blockDim.x + threadIdx.x;
    const int wave = gid >> 5;
    const int lane = threadIdx.x & 31;
    const int tm = wave / 3;                 // 0..15
    const int tn = wave % 3;                 // 0..2
    const int m0 = tm * 16, n0 = tn * 16;
    const int half = lane >> 4;
    const int l16  = lane & 15;
    const int kOff = half * 2;

    const int col  = n0 + l16;
    const int colC = col < NLAB ? col : (NLAB - 1);   // clamp to stay in-bounds

    const float* Ap = Hin + (size_t)(m0 + l16) * H_SZ + kOff;
    const float* Bp = W2  + (size_t)kOff * NLAB + colC;

    v8f c0 = {}; v8f c1 = {};
    #pragma unroll 2
    for (int k = 0; k < H_SZ; k += 8) {
        v2f a0 = *(const v2f*)(Ap + k);
        v2f b0; b0.x = Bp[k * NLAB];       b0.y = Bp[(k + 1) * NLAB];
        c0 = __builtin_amdgcn_wmma_f32_16x16x4_f32(false, a0, false, b0,
                                                   (short)0, c0, false, false);
        v2f a1 = *(const v2f*)(Ap + k + 4);
        v2f b1; b1.x = Bp[(k + 4) * NLAB]; b1.y = Bp[(k + 5) * NLAB];
        c1 = __builtin_amdgcn_wmma_f32_16x16x4_f32(false, a1, false, b1,
                                                   (short)0, c1, false, false);
    }

    if (col < NLAB) {
        const float bv    = bias2[col];
        const int   rbase = m0 + half * 8;
        #pragma unroll
        for (int r = 0; r < 8; ++r) {
            Out[(size_t)(rbase + r) * NLAB + col] = c0[r] + c1[r] + bv;
        }
    }
}

// ---------------------------------------------------------------------------
// Host-side launch
// ---------------------------------------------------------------------------
extern "C" void kernel_launch(void* const* d_in, const int* in_sizes, int n_in,
                              void* d_out, int out_size, void* d_ws, size_t ws_size,
                              hipStream_t stream) {
    const float* seq   = (const float*)d_in[0];   // [256,512,768]
    const int*   e1p   = (const int*)  d_in[1];   // [256]
    const int*   e2p   = (const int*)  d_in[2];   // [256]
    const int*   l1    = (const int*)  d_in[3];   // [256]
    const int*   l2    = (const int*)  d_in[4];   // [256]
    const float* wemb  = (const float*)d_in[5];   // [11,25]
    const float* W1    = (const float*)d_in[6];   // [2354,768]
    const float* b1    = (const float*)d_in[7];   // [768]
    const float* W2    = (const float*)d_in[8];   // [768,42]
    const float* b2    = (const float*)d_in[9];   // [42]
    float*       out   = (float*)d_out;           // [256,42]

    // workspace: combined [256,2368] f32, then h [256,768] f32
    float* comb = (float*)d_ws;
    float* hbuf = comb + (size_t)B_SZ * CINP;

    // 1) feature assembly (HBM-bound gather + masked mean)
    build_combined<<<B_SZ, 256, 0, stream>>>(seq, e1p, e2p, l1, l2, wemb, comb);

    // 2) GEMM1 + ReLU: 768 tiles, 1 wave each -> 96 blocks x 256 threads
    gemm1_relu<<<(16 * 48 * 32) / 256, 256, 0, stream>>>(comb, W1, b1, hbuf);

    // 3) GEMM2: 48 tiles, 1 wave each -> 12 blocks x 128 threads
    gemm2<<<(16 * 3 * 32) / 128, 128, 0, stream>>>(hbuf, W2, b2, out);
}